// Block2x2DiagProductBmm_7954279432745
// MI455X (gfx1250) — compile-verified
//
#include <hip/hip_runtime.h>
#include <stdint.h>

// ---------------------------------------------------------------------------
// Learned butterfly transform, SIZE=1024, BATCH=16384, fp32.
// Memory-bound: 134 MB HBM -> ~5.8us floor at 23.3 TB/s. Single LDS pass:
//   phase 0: GLOBAL_LOAD_ASYNC_TO_LDS_B128 (ASYNCcnt) global -> LDS
//   phase 1: stages 0-3 fused into one 16x16 matrix -> V_WMMA_F32_16X16X4_F32
//   phase 2: stages 4-7 fused into 16 matrices      -> V_WMMA_F32_16X16X4_F32
//   phase 3: stages 8-9 scalar butterflies in LDS
//   phase 4: GLOBAL_STORE_ASYNC_FROM_LDS_B128 LDS -> global
// Natural-order indexing (z[n]=y[bitrev(n)]) removes both bit-reversal copies:
// stage i pairs (n, n+2^i) with matrix j = bitrev_i(n mod 2^i).
// All loops have uniform trip counts -> EXEC all-ones (WMMA requirement).
// ---------------------------------------------------------------------------

#define SIZEN 1024
#define BT    32            // batch columns per workgroup
#define PITCH 1028          // floats per LDS column (16B-aligned, ~2-way banks)
#define TPB   256           // 8 waves (wave32)

typedef float v2f __attribute__((ext_vector_type(2)));
typedef float v4f __attribute__((ext_vector_type(4)));
typedef float v8f __attribute__((ext_vector_type(8)));

#define AS1 __attribute__((address_space(1)))
#define AS3 __attribute__((address_space(3)))

// Async-to-LDS builtins take GCC-style int4 pointers (per hipcc diagnostic:
// "__attribute__((__vector_size__(4*sizeof(int)))) int __device__ *").
typedef int b128_t __attribute__((vector_size(16)));

#if defined(__has_builtin)
#  if __has_builtin(__builtin_amdgcn_global_load_async_to_lds_b128) && \
      __has_builtin(__builtin_amdgcn_global_store_async_from_lds_b128)
#    define HAVE_ASYNC_LDS 1
#  endif
#endif
#ifndef HAVE_ASYNC_LDS
#  define HAVE_ASYNC_LDS 0
#endif

__device__ __forceinline__ void wait_async0() {
#if defined(__has_builtin) && __has_builtin(__builtin_amdgcn_s_wait_asynccnt)
  __builtin_amdgcn_s_wait_asynccnt(0);
#else
  asm volatile("s_wait_asynccnt 0x0" ::: "memory");
#endif
}

struct AB { const float* p[10]; };   // stage i: 2^i matrices of 2x2, row-major

__device__ __forceinline__ int brevN(unsigned x, int bits) {
  return bits ? (int)(__brev(x) >> (32 - bits)) : 0;
}

// ---------------------------------------------------------------------------
// Pre-kernel: compose 4 butterfly stages into dense 16x16 matrices.
//   mat 0     -> G0 (stages 0-3), shared by all 64 contiguous 16-row blocks
//   mat 1..16 -> G1[lo4] (stages 4-7), selected by n mod 16
// ---------------------------------------------------------------------------
__global__ void compose_g_kernel(AB ab, float* __restrict__ G) {
  const int mat  = blockIdx.x;
  const int lane = threadIdx.x;
  if (lane >= 16) return;
  float g[16];
#pragma unroll
  for (int r = 0; r < 16; ++r) g[r] = (r == lane) ? 1.0f : 0.0f;
  const int group = (mat == 0) ? 0 : 1;
  const int lo4   = (mat == 0) ? 0 : (mat - 1);
#pragma unroll
  for (int b = 0; b < 4; ++b) {
    const int stage = group * 4 + b;
    const float* M = ab.p[stage];
#pragma unroll
    for (int t0 = 0; t0 < 16; ++t0) {
      if (t0 & (1 << b)) continue;
      const int t1 = t0 | (1 << b);
      const int tl = t0 & ((1 << b) - 1);
      const int nlow = (group == 0) ? tl : (lo4 | (tl << 4));
      const int j = brevN((unsigned)nlow, stage);
      const float A = M[4*j+0], B = M[4*j+1], C = M[4*j+2], D = M[4*j+3];
      const float y0 = g[t0], y1 = g[t1];
      g[t0] = fmaf(A, y0, B * y1);
      g[t1] = fmaf(C, y0, D * y1);
    }
  }
#pragma unroll
  for (int r = 0; r < 16; ++r) G[mat*256 + r*16 + lane] = g[r];
}

// ---------------------------------------------------------------------------
// Main kernel: one workgroup = 32 batch columns x full 1024 rows in LDS.
// LDS layout: lds[col*PITCH + n]  (column-major; async b128 both directions).
// ---------------------------------------------------------------------------
__global__ __launch_bounds__(TPB) void butterfly_wmma_kernel(
    const float* __restrict__ x, AB ab, const float* __restrict__ G,
    float* __restrict__ out) {
  extern __shared__ float lds[];            // BT*PITCH*4 = 131,584 B
  const int tid  = threadIdx.x;
  const int lane = tid & 31;
  const int w    = tid >> 5;
  const int b0   = blockIdx.x * BT;

  // ---- Phase 0: global -> LDS, 16B per lane per op, contiguous both sides.
  float* xg = (float*)(x + (size_t)b0 * SIZEN);   // const dropped for builtin
#pragma unroll 4
  for (int rr = 0; rr < (BT * SIZEN / 4) / TPB; ++rr) {   // 32 uniform iters
    const int it  = tid + rr * TPB;
    const int col = it >> 8;                // SIZEN/4 == 256
    const int s4  = it & 255;
#if HAVE_ASYNC_LDS
    __builtin_amdgcn_global_load_async_to_lds_b128(
        (AS1 b128_t*)(xg + (size_t)col * SIZEN + s4 * 4),
        (AS3 b128_t*)((AS3 float*)lds + col * PITCH + s4 * 4), 0, 0);
#else
    const v4f v = ((const v4f*)xg)[col * 256 + s4];
    *(v4f*)&lds[col * PITCH + s4 * 4] = v;
#endif
  }
#if HAVE_ASYNC_LDS
  wait_async0();
#endif
  __syncthreads();

  // WMMA f32 16x16x4 per-lane operand coordinates (ISA 7.12.2):
  //   A (16x4): m=lane&15; VGPR0/1 = A[m][K]/A[m][K+1], K = 0 (lanes 0-15)
  //             or 2 (lanes 16-31).  B (4x16) mirrored.  D: VGPR v -> row
  //             v (lanes 0-15) / v+8 (lanes 16-31), col = lane&15.
  const int m16  = lane & 15;
  const int hi   = lane >> 4;
  const int offA = hi * 2;

  // ---- Phase 1: stages 0-3, one shared matrix G0, 64 contiguous blocks.
  v2f a0[4];
#pragma unroll
  for (int k = 0; k < 4; ++k) {
    a0[k].x = G[m16*16 + 4*k + offA + 0];
    a0[k].y = G[m16*16 + 4*k + offA + 1];
  }
  for (int rr = 0; rr < 16; ++rr) {         // uniform: 128 tiles / 8 waves
    const int T   = w + rr * 8;
    const int blk = T >> 1;
    const int c   = ((T & 1) << 4) + m16;
    v8f acc = {0.f, 0.f, 0.f, 0.f, 0.f, 0.f, 0.f, 0.f};
#pragma unroll
    for (int k = 0; k < 4; ++k) {
      // B chunk rows are contiguous in n -> single 8B LDS read per lane.
      const v2f bb = *(const v2f*)&lds[c * PITCH + blk * 16 + 4*k + offA];
      acc = __builtin_amdgcn_wmma_f32_16x16x4_f32(false, a0[k], false, bb,
                                                  (short)0, acc, false, false);
    }
    // D rows contiguous in n -> two 16B LDS writes per lane.
    float* dp = &lds[c * PITCH + blk * 16 + hi * 8];
    *(v4f*)(dp + 0) = (v4f){acc[0], acc[1], acc[2], acc[3]};
    *(v4f*)(dp + 4) = (v4f){acc[4], acc[5], acc[6], acc[7]};
  }
  __syncthreads();

  // ---- Phase 2: stages 4-7. Block beta: rows n = (beta&15) + t*16 +
  //      (beta>>4)*256, matrix G1[beta&15].
  for (int rr = 0; rr < 16; ++rr) {
    const int T    = w + rr * 8;
    const int beta = T >> 1;
    const int lo4  = beta & 15;
    const int hi2  = beta >> 4;
    const int c    = ((T & 1) << 4) + m16;
    const int base = lo4 + hi2 * 256;
    const float* Gb = G + (1 + lo4) * 256;
    v2f a1[4];
#pragma unroll
    for (int k = 0; k < 4; ++k) {
      a1[k].x = Gb[m16*16 + 4*k + offA + 0];
      a1[k].y = Gb[m16*16 + 4*k + offA + 1];
    }
    v8f acc = {0.f, 0.f, 0.f, 0.f, 0.f, 0.f, 0.f, 0.f};
#pragma unroll
    for (int k = 0; k < 4; ++k) {
      const int t0 = 4*k + offA;
      v2f bb;
      bb.x = lds[c * PITCH + base + (t0 + 0) * 16];
      bb.y = lds[c * PITCH + base + (t0 + 1) * 16];
      acc = __builtin_amdgcn_wmma_f32_16x16x4_f32(false, a1[k], false, bb,
                                                  (short)0, acc, false, false);
    }
#pragma unroll
    for (int v = 0; v < 8; ++v)
      lds[c * PITCH + base + (hi * 8 + v) * 16] = acc[v];
  }
  __syncthreads();

  // ---- Phase 3: stages 8 and 9, scalar butterflies.
  //      Lanes take consecutive n at fixed col -> conflict-free DS.
#pragma unroll
  for (int i = 8; i <= 9; ++i) {
    const float* M = ab.p[i];
    const int st = 1 << i;
    for (int rr = 0; rr < (512 * BT) / TPB; ++rr) {   // 64 uniform iters
      const int idx = tid + rr * TPB;
      const int col = idx >> 9;
      const int p   = idx & 511;
      const int n0  = ((p >> i) << (i + 1)) | (p & (st - 1));
      const int n1  = n0 + st;
      const int j   = brevN((unsigned)(n0 & (st - 1)), i);
      const float A = M[4*j+0], Bv = M[4*j+1], C = M[4*j+2], D = M[4*j+3];
      float* q = &lds[col * PITCH];
      const float y0 = q[n0];
      const float y1 = q[n1];
      q[n0] = fmaf(A, y0, Bv * y1);
      q[n1] = fmaf(C, y0, D  * y1);
    }
    __syncthreads();
  }

  // ---- Phase 4: LDS -> global, 16B per lane per op, contiguous both sides.
  float* og = out + (size_t)b0 * SIZEN;
#pragma unroll 4
  for (int rr = 0; rr < (BT * SIZEN / 4) / TPB; ++rr) {
    const int it  = tid + rr * TPB;
    const int col = it >> 8;
    const int s4  = it & 255;
#if HAVE_ASYNC_LDS
    __builtin_amdgcn_global_store_async_from_lds_b128(
        (AS1 b128_t*)(og + (size_t)col * SIZEN + s4 * 4),
        (AS3 b128_t*)((AS3 float*)lds + col * PITCH + s4 * 4), 0, 0);
#else
    ((v4f*)og)[col * 256 + s4] = *(const v4f*)&lds[col * PITCH + s4 * 4];
#endif
  }
#if HAVE_ASYNC_LDS
  wait_async0();    // stores complete before wave retires
#endif
}

extern "C" void kernel_launch(void* const* d_in, const int* in_sizes, int n_in,
                              void* d_out, int out_size, void* d_ws, size_t ws_size,
                              hipStream_t stream) {
  (void)n_in; (void)out_size; (void)ws_size;
  const float* x = (const float*)d_in[0];
  AB ab;
  for (int i = 0; i < 10; ++i) ab.p[i] = (const float*)d_in[1 + i];
  float* G = (float*)d_ws;                      // 17 * 256 floats = 17,408 B
  const int batch = in_sizes[0] / SIZEN;        // 16384

  compose_g_kernel<<<dim3(17), dim3(32), 0, stream>>>(ab, G);

  butterfly_wmma_kernel<<<dim3(batch / BT), dim3(TPB),
                          BT * PITCH * sizeof(float), stream>>>(
      x, ab, G, (float*)d_out);
}